// DenseAtt_1546188227118
// MI455X (gfx1250) — compile-verified
//
#include <hip/hip_runtime.h>

typedef float v2f  __attribute__((ext_vector_type(2)));
typedef float v4f  __attribute__((ext_vector_type(4)));
typedef float v8f  __attribute__((ext_vector_type(8)));

// ---------------------------------------------------------------------------
// Phase 1: left[i] = x[i,:] @ W[0:F] + b ; right[i] = x[i,:] @ W[F:2F]
// One wave (32 lanes) computes 16 rows via V_WMMA_F32_16X16X4_F32.
// A tile (16x4 f32): lanes 0-15 hold K=0,1 ; lanes 16-31 hold K=2,3.
// B tile (4x16 f32): W value broadcast across all 16 columns, so every
// column of the C tile equals the exact f32 dot product.
// ---------------------------------------------------------------------------
__global__ __launch_bounds__(32)
void denseatt_gemv_wmma(const float* __restrict__ x,
                        const float* __restrict__ W,
                        const float* __restrict__ bias,
                        float* __restrict__ left,
                        float* __restrict__ right,
                        int F) {
  const int m0    = blockIdx.x * 16;
  const int lane  = threadIdx.x;          // 0..31
  const int row   = m0 + (lane & 15);
  const int khalf = (lane >> 4) * 2;      // 0 for lanes 0-15, 2 for lanes 16-31

  const float* xrow = x + (long long)row * F;

  v8f accL = {};
  v8f accR = {};

  #pragma unroll 4
  for (int k = 0; k < F; k += 4) {
    v2f a, bl, br;
    a.x  = xrow[k + khalf + 0];
    a.y  = xrow[k + khalf + 1];
    bl.x = W[k + khalf + 0];            // left half of W
    bl.y = W[k + khalf + 1];
    br.x = W[F + k + khalf + 0];        // right half of W
    br.y = W[F + k + khalf + 1];
    // D = A*B + C ; 8 args: (neg_a, A, neg_b, B, c_mod, C, reuse_a, reuse_b)
    accL = __builtin_amdgcn_wmma_f32_16x16x4_f32(false, a, false, bl,
                                                 (short)0, accL, false, false);
    accR = __builtin_amdgcn_wmma_f32_16x16x4_f32(false, a, false, br,
                                                 (short)0, accR, false, false);
  }

  // C layout: VGPR v holds row m0+v (lanes 0-15) and row m0+8+v (lanes 16-31);
  // all 16 columns identical -> lane 0 and lane 16 write the results.
  const float bv = bias[0];
  if ((lane & 15) == 0) {
    const int base = m0 + ((lane >> 4) << 3);
    #pragma unroll
    for (int v = 0; v < 8; ++v) {
      left[base + v]  = accL[v] + bv;   // fold bias into left
      right[base + v] = accR[v];
    }
  }
}

// ---------------------------------------------------------------------------
// Phase 2: out[i,j] = adj[i,j] * sigmoid(left[i] + right[j])
// Bandwidth-bound: 512 MB of streaming traffic -> ~22 us at 23.3 TB/s.
// b128 vector access, non-temporal policy on the streamed tensors so the
// reused 32 KB left/right vectors stay resident in cache.
// ---------------------------------------------------------------------------
__global__ __launch_bounds__(256)
void denseatt_fuse(const float* __restrict__ adj,
                   const float* __restrict__ left,
                   const float* __restrict__ right,
                   float* __restrict__ out,
                   int N) {
  const int i  = blockIdx.y;                                   // row (uniform)
  const int j4 = (blockIdx.x * blockDim.x + threadIdx.x) << 2; // 4 floats/thread
  const long long off = (long long)i * N + j4;

  const float l = left[i];                       // wave-uniform -> s_load
  const v4f  r = *(const v4f*)(right + j4);      // cached, reused N times
  const v4f  a = __builtin_nontemporal_load((const v4f*)(adj + off));

  const float nlog2e = -1.44269504088896340736f;
  v4f o;
  #pragma unroll
  for (int c = 0; c < 4; ++c) {
    const float s = l + r[c];
    const float e = __builtin_amdgcn_exp2f(s * nlog2e);   // exp(-s)
    o[c] = a[c] * __builtin_amdgcn_rcpf(1.0f + e);        // adj * sigmoid(s)
  }
  __builtin_nontemporal_store(o, (v4f*)(out + off));
}

// ---------------------------------------------------------------------------
// inputs (setup_inputs order): x [N*F], adj [N*N], W [2F], b [1]
// ---------------------------------------------------------------------------
extern "C" void kernel_launch(void* const* d_in, const int* in_sizes, int n_in,
                              void* d_out, int out_size, void* d_ws, size_t ws_size,
                              hipStream_t stream) {
  const float* x   = (const float*)d_in[0];
  const float* adj = (const float*)d_in[1];
  const float* W   = (const float*)d_in[2];
  const float* b   = (const float*)d_in[3];
  float* out = (float*)d_out;

  const int F = in_sizes[2] / 2;   // 256
  const int N = in_sizes[0] / F;   // 8192

  float* left  = (float*)d_ws;
  float* right = left + N;

  // Phase 1: 512 waves, each producing 16 rows of left/right via WMMA.
  denseatt_gemv_wmma<<<N / 16, 32, 0, stream>>>(x, W, b, left, right, F);

  // Phase 2: one float4 per thread; grid (N/1024, N).
  dim3 grid(N / (256 * 4), N);
  denseatt_fuse<<<grid, 256, 0, stream>>>(adj, left, right, out, N);
}